// HMP_MACELayer_40329742909548
// MI455X (gfx1250) — compile-verified
//
#include <hip/hip_runtime.h>
#include <hip/hip_bf16.h>
#include <math.h>

// ---------------- problem constants ----------------
constexpr int N_   = 16384;
constexpr int E_   = 131072;
constexpr int F_   = 128;
constexpr int L_   = 9;
constexpr int R_   = 8;
constexpr int DOUT_= L_ * F_;   // 1152
constexpr int K_   = 1024;
constexpr int HID_ = 64;
constexpr int DA_  = 64;
constexpr float EPS_ = 1e-9f;

typedef float v2f __attribute__((ext_vector_type(2)));
typedef float v8f __attribute__((ext_vector_type(8)));
typedef unsigned int v4u __attribute__((ext_vector_type(4)));
typedef int v4i __attribute__((ext_vector_type(4)));
typedef int v8i __attribute__((ext_vector_type(8)));

#define WMMA_F32(a, b, c) \
    __builtin_amdgcn_wmma_f32_16x16x4_f32(false, (a), false, (b), (short)0, (c), false, false)

// =====================================================================
// Wide fp32 WMMA GEMM (direct global loads): each wave computes a 16x64
// tile with 4 accumulators sharing one A fragment (4x A reuse).
// Grid exact: tiles = (M/16)*(Nt/64), 4 waves (128 thr) per block.
// =====================================================================
__global__ void gemm_wmma_f32(const float* __restrict__ A, int lda,
                              const float* __restrict__ B, int ldb,
                              const float* __restrict__ Cadd, int ldc_add, int addW,
                              float* __restrict__ D, int ldd,
                              int M, int Nt, int Kt) {
    const int lane  = threadIdx.x & 31;
    const int wave  = threadIdx.x >> 5;
    const int tilesN = Nt >> 6;               // 64-wide super-tiles
    const int tid = blockIdx.x * (blockDim.x >> 5) + wave;
    const int bm = tid / tilesN;
    const int bn = tid - bm * tilesN;
    if (bm * 16 >= M) return;                 // wave-uniform (EXEC stays full)

    const int row = lane & 15;                // M-row for A, N-col for B/C/D
    const int grp = lane >> 4;                // lane-group: K (+0/+2), M (+0/+8)

    const float* __restrict__ Arow = A + (size_t)(bm * 16 + row) * lda;
    const float* __restrict__ B0   = B + bn * 64 + row;

    v8f acc0 = {}, acc1 = {}, acc2 = {}, acc3 = {};
    for (int k = 0; k < Kt; k += 4) {
        v2f a;
        a.x = Arow[k + grp * 2 + 0];
        a.y = Arow[k + grp * 2 + 1];
        const float* Bk0 = B0 + (size_t)(k + grp * 2) * ldb;
        const float* Bk1 = Bk0 + ldb;
        v2f b0, b1, b2, b3;
        b0.x = Bk0[0];  b0.y = Bk1[0];
        b1.x = Bk0[16]; b1.y = Bk1[16];
        b2.x = Bk0[32]; b2.y = Bk1[32];
        b3.x = Bk0[48]; b3.y = Bk1[48];
        acc0 = WMMA_F32(a, b0, acc0);
        acc1 = WMMA_F32(a, b1, acc1);
        acc2 = WMMA_F32(a, b2, acc2);
        acc3 = WMMA_F32(a, b3, acc3);
    }

    const int colBase = bn * 64 + row;
    #pragma unroll
    for (int v = 0; v < 8; ++v) {
        const int r = bm * 16 + v + 8 * grp;
        float o0 = acc0[v], o1 = acc1[v], o2 = acc2[v], o3 = acc3[v];
        if (Cadd) {
            const float* cr = Cadd + (size_t)r * ldc_add;
            if (colBase +  0 < addW) o0 += cr[colBase +  0];
            if (colBase + 16 < addW) o1 += cr[colBase + 16];
            if (colBase + 32 < addW) o2 += cr[colBase + 32];
            if (colBase + 48 < addW) o3 += cr[colBase + 48];
        }
        float* dr = D + (size_t)r * ldd;
        dr[colBase +  0] = o0;
        dr[colBase + 16] = o1;
        dr[colBase + 32] = o2;
        dr[colBase + 48] = o3;
    }
}

// =====================================================================
// TDM-staged fp32 WMMA GEMM for the big 1152-wide products.
// Block = 256 thr (8 waves) computes 64x128; B panel (32xKc rows x 128
// cols, 16 KB) is DMA'd into LDS by the Tensor Data Mover (D# built per
// ISA ch.8), waves feed WMMAs from LDS while A streams from global/L2.
// Requires M%64==0, Nt%128==0, Kt%32==0.
// =====================================================================
__global__ void gemm_wmma_tdm(const float* __restrict__ A, int lda,
                              const float* __restrict__ B, int ldb,
                              const float* __restrict__ Cadd, int ldc_add, int addW,
                              float* __restrict__ D, int ldd,
                              int M, int Nt, int Kt) {
    __shared__ float Bs[32 * 128];            // one K-chunk of the B panel
    const int lane = threadIdx.x & 31;
    const int wave = threadIdx.x >> 5;
    const int nb  = Nt >> 7;                  // 128-wide N blocks
    const int bmB = blockIdx.x / nb;
    const int bnB = blockIdx.x - bmB * nb;
    const int wm  = wave & 3;                 // 4 wave-rows
    const int wn  = wave >> 2;                // 2 wave-cols (64 each)
    const int row = lane & 15;
    const int grp = lane >> 4;

    const float* __restrict__ Arow = A + (size_t)(bmB * 64 + wm * 16 + row) * lda;
    const unsigned lds_base = (unsigned)(size_t)(void*)&Bs[0];

    v8f acc0 = {}, acc1 = {}, acc2 = {}, acc3 = {};
    for (int k0 = 0; k0 < Kt; k0 += 32) {
        __syncthreads();                      // Bs reuse guard
        if (wave == 0) {
            // ---- Tensor DMA descriptor (D#) for 128x32 fp32 tile of B ----
            const float* gB = B + (size_t)k0 * ldb + bnB * 128;
            const unsigned long long ga = (unsigned long long)(size_t)gB;
            v4u g0; v8i g1; v4i g2 = {0, 0, 0, 0}, g3 = {0, 0, 0, 0};
            g0[0] = 1u;                                     // count=1 (valid)
            g0[1] = lds_base;                               // lds_addr
            g0[2] = (unsigned)(ga & 0xFFFFFFFFu);           // global_addr lo
            g0[3] = (unsigned)((ga >> 32) & 0x01FFFFFFu)    // global_addr hi
                  | (2u << 30);                             // type=2 (image)
            const unsigned td0 = (unsigned)Nt;              // tensor_dim0
            const unsigned td1 = (unsigned)Kt;              // tensor_dim1
            const unsigned long long s0 = (unsigned long long)ldb;
            g1[0] = (int)(2u << 16);                        // data_size=4B
            g1[1] = (int)((td0 & 0xFFFFu) << 16);           // dim0[15:0]
            g1[2] = (int)((td0 >> 16) | ((td1 & 0xFFFFu) << 16));
            g1[3] = (int)((td1 >> 16) | (128u << 16));      // tile_dim0=128
            g1[4] = 32;                                     // tile_dim1=32
            g1[5] = (int)(unsigned)(s0 & 0xFFFFFFFFu);      // dim0_stride lo
            g1[6] = (int)(unsigned)((s0 >> 32) & 0xFFFFu);  // dim0_stride hi
            g1[7] = 0;
#if defined(__clang_major__) && (__clang_major__ >= 23)
            v8i g4 = {0, 0, 0, 0, 0, 0, 0, 0};
            __builtin_amdgcn_tensor_load_to_lds(g0, g1, g2, g3, g4, 0);
#else
            __builtin_amdgcn_tensor_load_to_lds(g0, g1, g2, g3, 0);
#endif
            __builtin_amdgcn_s_wait_tensorcnt(0);
        }
        __syncthreads();

        const float* __restrict__ Ab = Arow + k0;
        if (k0 + 32 < Kt) __builtin_prefetch(Arow + k0 + 32, 0, 3);

        #pragma unroll
        for (int kk = 0; kk < 32; kk += 4) {
            v2f a;
            a.x = Ab[kk + grp * 2 + 0];
            a.y = Ab[kk + grp * 2 + 1];
            const float* bs0 = &Bs[(kk + grp * 2) * 128 + wn * 64 + row];
            const float* bs1 = bs0 + 128;
            v2f b0, b1, b2, b3;
            b0.x = bs0[0];  b0.y = bs1[0];
            b1.x = bs0[16]; b1.y = bs1[16];
            b2.x = bs0[32]; b2.y = bs1[32];
            b3.x = bs0[48]; b3.y = bs1[48];
            acc0 = WMMA_F32(a, b0, acc0);
            acc1 = WMMA_F32(a, b1, acc1);
            acc2 = WMMA_F32(a, b2, acc2);
            acc3 = WMMA_F32(a, b3, acc3);
        }
    }

    const int colBase = bnB * 128 + wn * 64 + row;
    #pragma unroll
    for (int v = 0; v < 8; ++v) {
        const int r = bmB * 64 + wm * 16 + v + 8 * grp;
        float o0 = acc0[v], o1 = acc1[v], o2 = acc2[v], o3 = acc3[v];
        if (Cadd) {
            const float* cr = Cadd + (size_t)r * ldc_add;
            if (colBase +  0 < addW) o0 += cr[colBase +  0];
            if (colBase + 16 < addW) o1 += cr[colBase + 16];
            if (colBase + 32 < addW) o2 += cr[colBase + 32];
            if (colBase + 48 < addW) o3 += cr[colBase + 48];
        }
        float* dr = D + (size_t)r * ldd;
        dr[colBase +  0] = o0;
        dr[colBase + 16] = o1;
        dr[colBase + 32] = o2;
        dr[colBase + 48] = o3;
    }
}

// =====================================================================
// Per-edge radial mix + scatter-add into upd(N, L, F).
// =====================================================================
__global__ void edge_scatter(const float* __restrict__ hW,
                             const float* __restrict__ edge_feats,
                             const float* __restrict__ edge_sh,
                             const float* __restrict__ W_rad_l,
                             const int*   __restrict__ edge_index,
                             float* __restrict__ upd) {
    const int e = blockIdx.x * 8 + (threadIdx.x >> 5);
    const int lane = threadIdx.x & 31;
    if (e >= E_) return;
    const int src = edge_index[e];
    const int dst = edge_index[E_ + e];

    float ef[R_];
    #pragma unroll
    for (int r = 0; r < R_; ++r) ef[r] = edge_feats[e * R_ + r];
    float sh[L_];
    #pragma unroll
    for (int l = 0; l < L_; ++l) sh[l] = edge_sh[e * L_ + l];

    #pragma unroll
    for (int j = 0; j < 4; ++j) {
        const int f = lane + 32 * j;
        float rw = 0.f;
        #pragma unroll
        for (int r = 0; r < R_; ++r) rw = fmaf(ef[r], W_rad_l[r * F_ + f], rw);
        const float t = hW[(size_t)src * F_ + f] * rw;
        float* base = upd + (size_t)dst * DOUT_ + f;
        #pragma unroll
        for (int l = 0; l < L_; ++l)
            unsafeAtomicAdd(base + l * F_, sh[l] * t);   // global_atomic_add_f32
    }
}

// ===================== gate MLP =====================
__global__ void gate_mlp(const float* __restrict__ h_local,
                         const float* __restrict__ W1, const float* __restrict__ b1,
                         const float* __restrict__ W2, const float* __restrict__ b2,
                         float* __restrict__ m_out) {
    __shared__ float hs[F_];
    __shared__ float red[HID_];
    const int i = blockIdx.x, t = threadIdx.x;
    hs[t]      = h_local[(size_t)i * DOUT_ + t];
    hs[t + 64] = h_local[(size_t)i * DOUT_ + 64 + t];
    __syncthreads();
    float acc = b1[t];
    for (int f = 0; f < F_; ++f) acc = fmaf(hs[f], W1[f * HID_ + t], acc);
    red[t] = fmaxf(acc, 0.f) * W2[t];
    __syncthreads();
    for (int s = 32; s > 0; s >>= 1) {
        if (t < s) red[t] += red[t + s];
        __syncthreads();
    }
    if (t == 0) m_out[i] = 1.f / (1.f + __expf(-(red[0] + b2[0])));
}

// ===================== top-K via rank counting =====================
__global__ void rank_count(const float* __restrict__ m, int* __restrict__ flag) {
    __shared__ int red[256];
    const int i = blockIdx.x;
    const float mi = m[i];
    int c = 0;
    for (int j = threadIdx.x; j < N_; j += 256) {
        const float mj = m[j];
        c += (mj > mi) || (mj == mi && j < i);
    }
    red[threadIdx.x] = c;
    __syncthreads();
    for (int s = 128; s > 0; s >>= 1) {
        if (threadIdx.x < s) red[threadIdx.x] += red[threadIdx.x + s];
        __syncthreads();
    }
    if (threadIdx.x == 0) flag[i] = (red[0] < K_) ? 1 : 0;
}

__global__ void scan_masters(const int* __restrict__ flag,
                             int* __restrict__ master_idx,
                             int* __restrict__ rank,
                             int* __restrict__ master_out) {
    __shared__ int s[1024];
    const int t = threadIdx.x;
    const int base = t * 16;
    int lsum = 0;
    #pragma unroll
    for (int u = 0; u < 16; ++u) lsum += flag[base + u];
    s[t] = lsum;
    __syncthreads();
    for (int off = 1; off < 1024; off <<= 1) {
        const int v = s[t] + ((t >= off) ? s[t - off] : 0);
        __syncthreads();
        s[t] = v;
        __syncthreads();
    }
    int pos = s[t] - lsum;
    #pragma unroll
    for (int u = 0; u < 16; ++u) {
        const int i = base + u;
        if (flag[i]) {
            master_idx[pos] = i;
            master_out[pos] = i;
            rank[i] = pos;
            ++pos;
        } else {
            rank[i] = K_;
        }
    }
}

__global__ void build_adj(const int* __restrict__ edge_index,
                          const int* __restrict__ rank,
                          float* __restrict__ adj) {
    const int e = blockIdx.x * 256 + threadIdx.x;
    if (e >= E_) return;
    const int rs = rank[edge_index[e]];
    const int rd = rank[edge_index[E_ + e]];
    adj[(size_t)rs * (K_ + 1) + rd] = 1.0f;
}

__global__ void gather_masters(const int* __restrict__ master_idx,
                               const float* __restrict__ h_local,
                               const float* __restrict__ pos,
                               float* __restrict__ h_m,
                               float* __restrict__ pos_m) {
    const int kb = blockIdx.x;
    const int i = master_idx[kb];
    for (int g = threadIdx.x; g < DOUT_; g += 256)
        h_m[(size_t)kb * DOUT_ + g] = h_local[(size_t)i * DOUT_ + g];
    if (threadIdx.x < 3) pos_m[kb * 3 + threadIdx.x] = pos[i * 3 + threadIdx.x];
}

__global__ void attn_kernel(const float* __restrict__ q,
                            const float* __restrict__ kk,
                            float* __restrict__ Avirt) {
    const int idx = blockIdx.x * 256 + threadIdx.x;
    const int i = idx >> 10, j = idx & 1023;
    float d = 0.f;
    #pragma unroll 4
    for (int c = 0; c < DA_; ++c) d = fmaf(q[i * DA_ + c], kk[j * DA_ + c], d);
    const float a = 1.f / (1.f + __expf(-d * 0.125f));
    Avirt[idx] = (a > 0.5f) ? a : 0.f;
}

__global__ void arec_kernel(const float* __restrict__ Avirt,
                            const float* __restrict__ adj,
                            float* __restrict__ A_rec) {
    const int idx = blockIdx.x * 256 + threadIdx.x;
    const int i = idx >> 10, j = idx & 1023;
    float v = 0.f;
    if (i != j) {
        const float bv = (Avirt[j * K_ + i] > 0.f || Avirt[i * K_ + j] > 0.f) ? 1.f : 0.f;
        v = fminf(adj[(size_t)j * (K_ + 1) + i] + bv, 1.f);
    }
    A_rec[idx] = v;
}

// ===================== hierarchical geometric messages =====================
__global__ void hier_msg(const float* __restrict__ pos_m,
                         const float* __restrict__ hW_h,
                         const float* __restrict__ W_rad_h,
                         const float* __restrict__ A_rec,
                         float* __restrict__ upd_m) {
    const int i = blockIdx.x;
    const int f = threadIdx.x;
    __shared__ float pi3[3];
    if (f < 3) pi3[f] = pos_m[i * 3 + f];
    __syncthreads();
    const float px = pi3[0], py = pi3[1], pz = pi3[2];

    float wr[R_];
    #pragma unroll
    for (int n = 0; n < R_; ++n) wr[n] = W_rad_h[n * F_ + f];

    float acc[L_];
    #pragma unroll
    for (int l = 0; l < L_; ++l) acc[l] = 0.f;

    const float bes_scale = 0.63245553203f;          // sqrt(2/5)
    const float wpi = 0.62831853072f;                // pi / RCUT

    for (int j = 0; j < K_; ++j) {
        const float a = A_rec[(size_t)i * K_ + j];   // uniform across block
        if (a == 0.f) continue;
        const float vx = pos_m[j * 3 + 0] - px;
        const float vy = pos_m[j * 3 + 1] - py;
        const float vz = pos_m[j * 3 + 2] - pz;
        const float r = sqrtf(vx * vx + vy * vy + vz * vz);
        const float inv = 1.f / fmaxf(r, EPS_);
        const float ux = vx * inv, uy = vy * inv, uz = vz * inv;
        float sh[L_] = {1.f, ux, uy, uz, ux * uy, uy * uz,
                        3.f * uz * uz - 1.f, ux * uz, ux * ux - uy * uy};
        float rw = 0.f;
        #pragma unroll
        for (int n = 0; n < R_; ++n) {
            const float rad = bes_scale * __sinf((float)(n + 1) * wpi * r) * inv;
            rw = fmaf(rad, wr[n], rw);
        }
        const float tm = hW_h[(size_t)j * F_ + f] * rw * a;
        #pragma unroll
        for (int l = 0; l < L_; ++l) acc[l] = fmaf(sh[l], tm, acc[l]);
    }
    #pragma unroll
    for (int l = 0; l < L_; ++l)
        upd_m[(size_t)i * DOUT_ + l * F_ + f] = acc[l];
}

__global__ void final_blend(const float* __restrict__ h_local,
                            const float* __restrict__ h_hier,
                            const float* __restrict__ m,
                            const int* __restrict__ rank,
                            float* __restrict__ h_final) {
    const int idx = blockIdx.x * 256 + threadIdx.x;
    if (idx >= N_ * DOUT_) return;
    const int i = idx / DOUT_;
    const int g = idx - i * DOUT_;
    const float mi = m[i];
    float v = (1.f - mi) * h_local[idx];
    const int rk = rank[i];
    if (rk < K_) v = fmaf(mi, h_hier[(size_t)rk * DOUT_ + g], v);
    h_final[idx] = v;
}

// =====================================================================
extern "C" void kernel_launch(void* const* d_in, const int* in_sizes, int n_in,
                              void* d_out, int out_size, void* d_ws, size_t ws_size,
                              hipStream_t stream) {
    (void)in_sizes; (void)n_in; (void)out_size; (void)ws_size;

    const float* h          = (const float*)d_in[0];
    const float* pos        = (const float*)d_in[1];
    const float* edge_sh    = (const float*)d_in[2];
    const float* edge_feats = (const float*)d_in[3];
    const float* W_node_l   = (const float*)d_in[4];
    const float* W_rad_l    = (const float*)d_in[5];
    const float* W_prod_l   = (const float*)d_in[6];
    const float* ms_W1      = (const float*)d_in[7];
    const float* ms_b1      = (const float*)d_in[8];
    const float* ms_W2      = (const float*)d_in[9];
    const float* ms_b2      = (const float*)d_in[10];
    const float* vg_Wq      = (const float*)d_in[11];
    const float* vg_Wk      = (const float*)d_in[12];
    const float* W_node_h   = (const float*)d_in[13];
    const float* W_rad_h    = (const float*)d_in[14];
    const float* W_prod_h   = (const float*)d_in[15];
    const int*   edge_index = (const int*)d_in[16];

    // ---- output layout: h_final | A_virtual | m | master_idx(int bits) ----
    float* out      = (float*)d_out;
    float* h_final  = out;
    float* Avirt    = out + (size_t)N_ * DOUT_;
    float* m_out    = Avirt + (size_t)K_ * K_;
    int*   mast_out = (int*)(m_out + N_);

    // ---- workspace carve ----
    float* w = (float*)d_ws;
    float* hW      = w; w += (size_t)N_ * F_;
    float* upd     = w; w += (size_t)N_ * DOUT_;
    float* h_local = w; w += (size_t)N_ * DOUT_;
    float* adj     = w; w += (size_t)(K_ + 1) * (K_ + 1);
    float* h_m     = w; w += (size_t)K_ * DOUT_;
    float* pos_m   = w; w += (size_t)K_ * 3;
    float* qbuf    = w; w += (size_t)K_ * DA_;
    float* kbuf    = w; w += (size_t)K_ * DA_;
    float* A_rec   = w; w += (size_t)K_ * K_;
    float* hW_h    = w; w += (size_t)K_ * F_;
    float* upd_m   = w; w += (size_t)K_ * DOUT_;
    float* h_hier  = w; w += (size_t)K_ * DOUT_;
    int* iw = (int*)w;
    int* flag       = iw; iw += N_;
    int* master_idx = iw; iw += K_;
    int* rank       = iw; iw += N_;

    hipMemsetAsync(upd, 0, (size_t)N_ * DOUT_ * sizeof(float), stream);
    hipMemsetAsync(adj, 0, (size_t)(K_ + 1) * (K_ + 1) * sizeof(float), stream);

    // 1) hW = h @ W_node_l   (16384x128x128)
    gemm_wmma_f32<<<(N_/16)*(F_/64)/4, 128, 0, stream>>>(
        h, F_, W_node_l, F_, nullptr, 0, 0, hW, F_, N_, F_, F_);

    // 2) per-edge radial mix + scatter-add into upd
    edge_scatter<<<E_/8, 256, 0, stream>>>(hW, edge_feats, edge_sh, W_rad_l,
                                           edge_index, upd);

    // 3) h_local = upd @ W_prod_l + pad(h)   (16384x1152x1152, TDM-staged B)
    gemm_wmma_tdm<<<(N_/64)*(DOUT_/128), 256, 0, stream>>>(
        upd, DOUT_, W_prod_l, DOUT_, h, F_, F_, h_local, DOUT_, N_, DOUT_, DOUT_);

    // 4) gate MLP -> m
    gate_mlp<<<N_, HID_, 0, stream>>>(h_local, ms_W1, ms_b1, ms_W2, ms_b2, m_out);

    // 5) top-K selection
    rank_count<<<N_, 256, 0, stream>>>(m_out, flag);
    scan_masters<<<1, 1024, 0, stream>>>(flag, master_idx, rank, mast_out);

    // 6) adjacency over ranked edges
    build_adj<<<E_/256, 256, 0, stream>>>(edge_index, rank, adj);

    // 7) gather master rows
    gather_masters<<<K_, 256, 0, stream>>>(master_idx, h_local, pos, h_m, pos_m);

    // 8) q/k projections (first F cols of h_m)
    gemm_wmma_f32<<<(K_/16)*(DA_/64)/4, 128, 0, stream>>>(
        h_m, DOUT_, vg_Wq, DA_, nullptr, 0, 0, qbuf, DA_, K_, DA_, F_);
    gemm_wmma_f32<<<(K_/16)*(DA_/64)/4, 128, 0, stream>>>(
        h_m, DOUT_, vg_Wk, DA_, nullptr, 0, 0, kbuf, DA_, K_, DA_, F_);

    // 9) attention + receive matrix
    attn_kernel<<<(K_*K_)/256, 256, 0, stream>>>(qbuf, kbuf, Avirt);
    arec_kernel<<<(K_*K_)/256, 256, 0, stream>>>(Avirt, adj, A_rec);

    // 10) hW_h = h_m @ W_node_h   (1024x1152x128)
    gemm_wmma_f32<<<(K_/16)*(F_/64)/4, 128, 0, stream>>>(
        h_m, DOUT_, W_node_h, F_, nullptr, 0, 0, hW_h, F_, K_, F_, DOUT_);

    // 11) geometric hierarchical messages -> upd_m
    hier_msg<<<K_, F_, 0, stream>>>(pos_m, hW_h, W_rad_h, A_rec, upd_m);

    // 12) h_hier = upd_m @ W_prod_h + h_m   (1024x1152x1152, TDM-staged B)
    gemm_wmma_tdm<<<(K_/64)*(DOUT_/128), 256, 0, stream>>>(
        upd_m, DOUT_, W_prod_h, DOUT_, h_m, DOUT_, DOUT_, h_hier, DOUT_, K_, DOUT_, DOUT_);

    // 13) final blend -> h_final
    final_blend<<<(N_*DOUT_ + 255)/256, 256, 0, stream>>>(
        h_local, h_hier, m_out, rank, h_final);
}